// DeepPolySPUTransformer_39376260169760
// MI455X (gfx1250) — compile-verified
//
#include <hip/hip_runtime.h>

// Pure elementwise, memory-bound kernel (402 MB of traffic -> ~17.3 us floor
// at 23.3 TB/s). No matrix math exists in the reference ("diagonal matmul is
// elementwise"), so no WMMA: the gfx1250 path that matters here is wave32
// streaming with b128 non-temporal loads/stores + native v_exp/v_rcp VALU.

typedef float v4f __attribute__((ext_vector_type(4)));

#define SPU_ZERO 0.7071067811865476f  // sqrt(0.5)

__device__ __forceinline__ float fast_rcp(float v) {
    return __builtin_amdgcn_rcpf(v);  // v_rcp_f32
}

// spu(v) = v^2 - 0.5            for v >= 0
//        = sigmoid(-v) - 1      for v <  0   (== -1/(1+exp(-v)))
__device__ __forceinline__ float spu_f(float v) {
    float pos = __builtin_fmaf(v, v, -0.5f);
    float neg = -fast_rcp(1.0f + __expf(-v));   // v_exp_f32 + v_rcp_f32
    return (v >= 0.0f) ? pos : neg;
}

// dspu(v) = 2v                  for v >= 0
//         = -s(1-s), s=sigmoid(-v) = 1/(1+exp(v))   for v < 0
__device__ __forceinline__ float dspu_f(float v) {
    float s   = fast_rcp(1.0f + __expf(v));
    float neg = -s * (1.0f - s);
    float pos = v + v;
    return (v >= 0.0f) ? pos : neg;
}

__device__ __forceinline__ void spu_relax(float xx, float l, float u,
                                          float& o, float& nlo, float& nup) {
    float sl      = spu_f(l);
    float su      = spu_f(u);
    float slope   = (su - sl) * fast_rcp(u - l);
    float chord_b = __builtin_fmaf(-slope, l, sl);

    // case 2 (only selected when l >= 0  =>  a2 >= 0): tangent at midpoint.
    // dspu(a2) = 2*a2 ; spu(a2) - a2*dspu(a2) = -a2^2 - 0.5
    float a2  = 0.5f * (u + l);
    float wl2 = a2 + a2;
    float bl2 = __builtin_fmaf(-a2, a2, -0.5f);

    // case 3: a3 = max(a2, sqrt(0.5)) >= sqrt(0.5) > 0 -> same shortcut
    float a3  = fmaxf(a2, SPU_ZERO);
    float wl3 = a3 + a3;
    float bl3 = __builtin_fmaf(-a3, a3, -0.5f);

    // case 4 (l < 0, 0 < u < sqrt(0.5))
    bool  flat4 = sl > su;
    float wu4   = flat4 ? 0.0f : slope;
    float bu4   = flat4 ? fmaxf(sl, su) : chord_b;

    bool c1 = (u <= 0.0f);
    bool c2 = (l >= 0.0f);
    bool c3 = (u >= SPU_ZERO);

    // nested selects, innermost (c3) first
    float w_u = c3 ? slope   : wu4;
    float b_u = c3 ? chord_b : bu4;
    float w_l = c3 ? wl3     : 0.0f;
    float b_l = c3 ? bl3     : -0.5f;

    w_u = c2 ? (u + l)                       : w_u;
    b_u = c2 ? __builtin_fmaf(-u, l, 0.5f)   : b_u;   // -u*l + 0.5
    w_l = c2 ? wl2                           : w_l;
    b_l = c2 ? bl2                           : b_l;

    w_u = c1 ? 0.0f    : w_u;
    b_u = c1 ? sl      : b_u;
    w_l = c1 ? slope   : w_l;
    b_l = c1 ? chord_b : b_l;

    o   = spu_f(xx);
    nlo = __builtin_fmaf(w_l, l, b_l);
    nup = __builtin_fmaf(w_u, u, b_u);
}

__global__ __launch_bounds__(256)
void spu_vec4_kernel(const v4f* __restrict__ x,
                     const v4f* __restrict__ l,
                     const v4f* __restrict__ u,
                     v4f* __restrict__ o,
                     v4f* __restrict__ nl,
                     v4f* __restrict__ nu,
                     int nvec) {
    int i = blockIdx.x * blockDim.x + threadIdx.x;
    if (i >= nvec) return;

    // streaming: non-temporal b128 loads/stores (working set >> L2)
    v4f xv = __builtin_nontemporal_load(x + i);
    v4f lv = __builtin_nontemporal_load(l + i);
    v4f uv = __builtin_nontemporal_load(u + i);

    v4f ov, nlv, nuv;
#pragma unroll
    for (int k = 0; k < 4; ++k) {
        float oo, ll, uu;
        spu_relax(xv[k], lv[k], uv[k], oo, ll, uu);
        ov[k] = oo; nlv[k] = ll; nuv[k] = uu;
    }

    __builtin_nontemporal_store(ov,  o  + i);
    __builtin_nontemporal_store(nlv, nl + i);
    __builtin_nontemporal_store(nuv, nu + i);
}

__global__ __launch_bounds__(256)
void spu_scalar_kernel(const float* __restrict__ x,
                       const float* __restrict__ l,
                       const float* __restrict__ u,
                       float* __restrict__ o,
                       float* __restrict__ nl,
                       float* __restrict__ nu,
                       int n, int start) {
    int i = start + blockIdx.x * blockDim.x + threadIdx.x;
    if (i >= n) return;
    float oo, ll, uu;
    spu_relax(x[i], l[i], u[i], oo, ll, uu);
    __builtin_nontemporal_store(oo, o  + i);
    __builtin_nontemporal_store(ll, nl + i);
    __builtin_nontemporal_store(uu, nu + i);
}

extern "C" void kernel_launch(void* const* d_in, const int* in_sizes, int n_in,
                              void* d_out, int out_size, void* d_ws, size_t ws_size,
                              hipStream_t stream) {
    const float* x = (const float*)d_in[0];   // "x"
    const float* l = (const float*)d_in[1];   // "lower_bounds"
    const float* u = (const float*)d_in[2];   // "upper_bounds"
    float* out = (float*)d_out;               // [out | new_lower | new_upper]

    int n = in_sizes[0];
    if (n <= 0) return;

    float* nl = out + (size_t)n;
    float* nu = out + 2 * (size_t)n;

    const int block = 256;
    if ((n & 3) == 0) {
        // sections are float4-aligned (n % 4 == 0): full vectorized path
        int nvec = n >> 2;
        int grid = (nvec + block - 1) / block;
        spu_vec4_kernel<<<grid, block, 0, stream>>>(
            (const v4f*)x, (const v4f*)l, (const v4f*)u,
            (v4f*)out, (v4f*)nl, (v4f*)nu, nvec);
    } else {
        // odd N: output sections lose 16B alignment -> scalar fallback
        int grid = (n + block - 1) / block;
        spu_scalar_kernel<<<grid, block, 0, stream>>>(x, l, u, out, nl, nu, n, 0);
    }
}